// FEA_22608707846907
// MI455X (gfx1250) — compile-verified
//
#include <hip/hip_runtime.h>
#include <hip/hip_bf16.h>

// ---------------------------------------------------------------------------
// Factored pipeline:
//   G  = (Xhat @ X^T) / sqrt(C)          (B,32,32)   -- gram_kernel
//   mm = G^T @ X                          (B,32,4096) -- mm_wmma_kernel (v_wmma_f32_16x16x4_f32)
//   y  = conv3x3(mm, w) + b                          -- conv_kernel (LDS weights)
//   stats = mean/var over (B,H,W) per ch             -- bnstat_kernel
//   p  = maxpool2(relu(bn(y)))            (B,32,32,32)-- bnpool_kernel
//   out= 2*bilinear_up(p) + 0.2*f                    -- upsample_kernel
// ---------------------------------------------------------------------------

typedef float v2f __attribute__((ext_vector_type(2)));
typedef float v8f __attribute__((ext_vector_type(8)));

#define BATCH 2
#define CH    32
#define HH    64
#define WW    64
#define NPIX  (HH * WW)            // 4096
#define RSQRT_C 0.17677669529663687f  // 1/sqrt(32)

// ---- Kernel 1: G[b,cp,c] = (1/sqrt(C)) * sum_n Xhat[b,cp,n] * X[b,c,n] ----
// One wave per output element; lane-strided loads are fully coalesced.
__global__ __launch_bounds__(256) void gram_kernel(const float* __restrict__ X,
                                                   const float* __restrict__ Xh,
                                                   float* __restrict__ G) {
    int wave = (blockIdx.x * blockDim.x + threadIdx.x) >> 5;
    int lane = threadIdx.x & 31;
    int b   = wave >> 10;          // 1024 outputs per batch
    int idx = wave & 1023;
    int cp  = idx >> 5;
    int c   = idx & 31;
    const float* xh = Xh + (size_t)(b * CH + cp) * NPIX;
    const float* xv = X  + (size_t)(b * CH + c ) * NPIX;
    float s = 0.f;
    for (int n = lane; n < NPIX; n += 32)
        s = fmaf(xh[n], xv[n], s);
    #pragma unroll
    for (int off = 16; off; off >>= 1)
        s += __shfl_xor(s, off, 32);
    if (lane == 0)
        G[(b * CH + cp) * CH + c] = s * RSQRT_C;
}

// ---- Kernel 2: mm = G^T @ X via V_WMMA_F32_16X16X4_F32 --------------------
// A = G^T (32x32 fp32), B = X (32x4096 fp32), K = 32 in steps of 4.
// One wave owns a 32(M) x 16(N) output block (two 16x16 accumulators).
// Layouts per CDNA5 ISA:
//   A 16x4 f32 : lanes 0-15 -> M=0..15, VGPR0={K,K+2 by half}, VGPR1={K+1,K+3}
//   C/D 16x16  : VGPR v -> M = v + 8*(lane>=16), N = lane&15
__global__ __launch_bounds__(128) void mm_wmma_kernel(const float* __restrict__ X,
                                                      const float* __restrict__ G,
                                                      float* __restrict__ MM) {
    int wave = (blockIdx.x * blockDim.x + threadIdx.x) >> 5;
    int lane = threadIdx.x & 31;
    int b    = wave >> 8;          // 256 N-tiles per batch
    int nt   = wave & 255;
    int ncol = nt << 4;
    int half = lane >> 4;
    int lm   = lane & 15;

    const float* Gb = G + b * CH * CH;
    const float* Xb = X + (size_t)b * CH * NPIX;

    v8f c0 = {0.f, 0.f, 0.f, 0.f, 0.f, 0.f, 0.f, 0.f};
    v8f c1 = {0.f, 0.f, 0.f, 0.f, 0.f, 0.f, 0.f, 0.f};

    #pragma unroll
    for (int kk = 0; kk < CH; kk += 4) {
        int k0 = kk + 2 * half;
        v2f a0, a1, bm;
        // A[m][k] = G[k][m] (A = G^T); M-tile 0 rows 0..15, M-tile 1 rows 16..31
        a0.x = Gb[(k0    ) * CH + lm];
        a0.y = Gb[(k0 + 1) * CH + lm];
        a1.x = Gb[(k0    ) * CH + 16 + lm];
        a1.y = Gb[(k0 + 1) * CH + 16 + lm];
        // B[k][n] = X[k][ncol + n]
        bm.x = Xb[(size_t)(k0    ) * NPIX + ncol + lm];
        bm.y = Xb[(size_t)(k0 + 1) * NPIX + ncol + lm];
        c0 = __builtin_amdgcn_wmma_f32_16x16x4_f32(false, a0, false, bm,
                                                   (short)0, c0, false, false);
        c1 = __builtin_amdgcn_wmma_f32_16x16x4_f32(false, a1, false, bm,
                                                   (short)0, c1, false, false);
    }

    float* MMb = MM + (size_t)b * CH * NPIX;
    #pragma unroll
    for (int v = 0; v < 8; ++v) {
        int row = v + 8 * half;
        MMb[(size_t)(row     ) * NPIX + ncol + lm] = c0[v];
        MMb[(size_t)(row + 16) * NPIX + ncol + lm] = c1[v];
    }
}

// ---- Kernel 3: y = conv3x3_same(mm, w) + bias ------------------------------
// One thread per output pixel; this block's 288 weights staged through LDS.
__global__ __launch_bounds__(256) void conv_kernel(const float* __restrict__ MM,
                                                   const float* __restrict__ Wgt,
                                                   const float* __restrict__ Bias,
                                                   float* __restrict__ Y) {
    __shared__ float sw[CH * 9];
    int gid = blockIdx.x * 256 + threadIdx.x;
    int w = gid & 63;
    int h = (gid >> 6) & 63;
    int o = (gid >> 12) & 31;      // constant within a block (4096 px per (b,o))
    int b = gid >> 17;
    for (int j = threadIdx.x; j < CH * 9; j += 256)
        sw[j] = Wgt[o * CH * 9 + j];
    __syncthreads();

    float acc = Bias[o];
    const float* mb = MM + (size_t)b * CH * NPIX;
    for (int i = 0; i < CH; ++i) {
        const float* p  = mb + (size_t)i * NPIX;
        const float* wr = sw + i * 9;
        #pragma unroll
        for (int dy = -1; dy <= 1; ++dy) {
            int hh = h + dy;
            if ((unsigned)hh >= (unsigned)HH) continue;
            #pragma unroll
            for (int dx = -1; dx <= 1; ++dx) {
                int ww2 = w + dx;
                if ((unsigned)ww2 >= (unsigned)WW) continue;
                acc = fmaf(p[hh * WW + ww2], wr[(dy + 1) * 3 + (dx + 1)], acc);
            }
        }
    }
    Y[gid] = acc;
}

// ---- Kernel 4: per-channel biased mean / rstd over (B,H,W) -----------------
// One block per channel; deterministic LDS tree reduction (no atomics, so the
// graph replays bit-identically).
__global__ __launch_bounds__(256) void bnstat_kernel(const float* __restrict__ Y,
                                                     float* __restrict__ Stats) {
    __shared__ float s1[256];
    __shared__ float s2[256];
    int c = blockIdx.x;
    float sum = 0.f, ssq = 0.f;
    for (int t = threadIdx.x; t < BATCH * NPIX; t += 256) {
        int b = t >> 12;           // NPIX = 4096
        int n = t & (NPIX - 1);
        float v = Y[(size_t)(b * CH + c) * NPIX + n];
        sum += v;
        ssq = fmaf(v, v, ssq);
    }
    s1[threadIdx.x] = sum;
    s2[threadIdx.x] = ssq;
    __syncthreads();
    for (int off = 128; off; off >>= 1) {
        if ((int)threadIdx.x < off) {
            s1[threadIdx.x] += s1[threadIdx.x + off];
            s2[threadIdx.x] += s2[threadIdx.x + off];
        }
        __syncthreads();
    }
    if (threadIdx.x == 0) {
        const float inv = 1.0f / (float)(BATCH * NPIX);
        float mean = s1[0] * inv;
        float var  = s2[0] * inv - mean * mean;   // biased, like jnp.var
        Stats[c]      = mean;
        Stats[CH + c] = rsqrtf(var + 1e-5f);
    }
}

// ---- Kernel 5: p = maxpool2(relu(bn(y))) -----------------------------------
__global__ __launch_bounds__(256) void bnpool_kernel(const float* __restrict__ Y,
                                                     const float* __restrict__ Stats,
                                                     const float* __restrict__ Gam,
                                                     const float* __restrict__ Bet,
                                                     float* __restrict__ P) {
    int gid = blockIdx.x * 256 + threadIdx.x;   // 65536 = 2*32*32*32
    int wp = gid & 31;
    int hp = (gid >> 5) & 31;
    int c  = (gid >> 10) & 31;
    int b  = gid >> 15;
    float mean = Stats[c];
    float rstd = Stats[CH + c];
    float g    = Gam[c];
    float be   = Bet[c];
    const float* yb = Y + (size_t)(b * CH + c) * NPIX + (hp * 2) * WW + wp * 2;
    float m = 0.f;  // relu outputs are >= 0, so 0 is a safe identity
    #pragma unroll
    for (int dy = 0; dy < 2; ++dy)
        #pragma unroll
        for (int dx = 0; dx < 2; ++dx) {
            float v = fmaf(g, (yb[dy * WW + dx] - mean) * rstd, be);
            m = fmaxf(m, fmaxf(v, 0.f));
        }
    P[gid] = m;
}

// ---- Kernel 6: out = 2 * bilinear_up_align(p) + 0.2 * f --------------------
__global__ __launch_bounds__(256) void upsample_kernel(const float* __restrict__ P,
                                                       const float* __restrict__ F,
                                                       float* __restrict__ Out) {
    int gid = blockIdx.x * 256 + threadIdx.x;   // 262144 = 2*32*64*64
    int ow = gid & 63;
    int oh = (gid >> 6) & 63;
    int c  = (gid >> 12) & 31;
    int b  = gid >> 17;
    const float s = 31.0f / 63.0f;              // (h-1)/(oh-1), align_corners
    float ys = oh * s, xs = ow * s;
    int y0 = (int)floorf(ys), x0 = (int)floorf(xs);
    int y1 = min(y0 + 1, 31),  x1 = min(x0 + 1, 31);
    float wy = ys - (float)y0, wx = xs - (float)x0;
    const float* pb = P + (size_t)(b * CH + c) * 1024;
    float top = pb[y0 * 32 + x0] * (1.f - wx) + pb[y0 * 32 + x1] * wx;
    float bot = pb[y1 * 32 + x0] * (1.f - wx) + pb[y1 * 32 + x1] * wx;
    float val = top * (1.f - wy) + bot * wy;
    Out[gid] = 2.0f * val + 0.2f * F[gid];
}

// ---------------------------------------------------------------------------
extern "C" void kernel_launch(void* const* d_in, const int* in_sizes, int n_in,
                              void* d_out, int out_size, void* d_ws, size_t ws_size,
                              hipStream_t stream) {
    const float* X    = (const float*)d_in[0];  // X_tnext      (2,32,64,64)
    const float* Xh   = (const float*)d_in[1];  // X_hat_tnext  (2,32,64,64)
    const float* F    = (const float*)d_in[2];  // f_tprev_t    (2,32,64,64)
    const float* Wgt  = (const float*)d_in[3];  // conv_w       (32,32,3,3)
    const float* Bias = (const float*)d_in[4];  // conv_b       (32,)
    const float* Gam  = (const float*)d_in[5];  // bn_gamma     (32,)
    const float* Bet  = (const float*)d_in[6];  // bn_beta      (32,)
    float* out = (float*)d_out;

    // Workspace layout (floats): G | MM | Y | Stats | P  (~2.4 MB total)
    float* ws    = (float*)d_ws;
    float* G     = ws;                               // 2048
    float* MM    = G  + BATCH * CH * CH;             // 262144
    float* Y     = MM + BATCH * CH * NPIX;           // 262144
    float* Stats = Y  + BATCH * CH * NPIX;           // 64 (mean[32], rstd[32])
    float* P     = Stats + 2 * CH;                   // 65536

    gram_kernel   <<<256,  256, 0, stream>>>(X, Xh, G);          // 2048 waves
    mm_wmma_kernel<<<128,  128, 0, stream>>>(X, G, MM);          // 512 waves
    conv_kernel   <<<1024, 256, 0, stream>>>(MM, Wgt, Bias, Y);
    bnstat_kernel <<<32,   256, 0, stream>>>(Y, Stats);
    bnpool_kernel <<<256,  256, 0, stream>>>(Y, Stats, Gam, Bet, P);
    upsample_kernel<<<1024, 256, 0, stream>>>(P, F, out);
}